// GNNEncoder_76278619177362
// MI455X (gfx1250) — compile-verified
//
#include <hip/hip_runtime.h>

typedef __attribute__((ext_vector_type(2))) float v2f;
typedef __attribute__((ext_vector_type(8))) float v8f;

#define NN 100000
#define NE 1600000
#define NG 128
#define BN_EPS 1e-5f

// ---------------- elementwise helpers ----------------

__global__ void copy_f4(const float4* __restrict__ src, float4* __restrict__ dst, int n4) {
  int i = blockIdx.x * blockDim.x + threadIdx.x;
  if (i < n4) dst[i] = src[i];
}

__global__ void zero_f(float* __restrict__ p, int n) {
  int i = blockIdx.x * blockDim.x + threadIdx.x;
  if (i < n) p[i] = 0.0f;
}

// ---------------- GIN aggregation: agg[dst] += x[src] ----------------
// agg is pre-initialized with x (the "+x_i" self term). 16 threads per edge,
// each handles a float4 chunk of the 64-wide feature row.
__global__ void scatter_add(const float* __restrict__ x, const int* __restrict__ srcI,
                            const int* __restrict__ dstI, float* __restrict__ agg) {
  long long tid = (long long)blockIdx.x * blockDim.x + threadIdx.x;
  if (tid >= (long long)NE * 16) return;
  int chunk = (int)(tid & 15);
  int e     = (int)(tid >> 4);
  int s = srcI[e];
  int d = dstI[e];
  const float4 v = *(const float4*)(x + (long long)s * 64 + chunk * 4);
  float* o = agg + (long long)d * 64 + chunk * 4;
  atomicAdd(o + 0, v.x);
  atomicAdd(o + 1, v.y);
  atomicAdd(o + 2, v.z);
  atomicAdd(o + 3, v.w);
}

// ---------------- WMMA f32 GEMM: out = act(in[M,K=64] @ W[64,Nout] + bias) --------
// Weights are staged ONCE per block into LDS in B-fragment order, so each WMMA's
// B operand is a single conflict-free ds_load_b64 (lane*8 + immediate offset).
// Per tile: one clause of 16 global_load_b64 for A fragments, then 16*NT
// v_wmma_f32_16x16x4_f32 fed from LDS.
// Fragment layouts per CDNA5 ISA:
//   A 16x4 f32: lane%16 = row M, lane/16 selects K pair (k,k+1)/(k+2,k+3)
//   B 4x16 f32: lane%16 = col N, lane/16 selects K pair
//   C/D 16x16:  VGPR v, row = v + (lane/16)*8, col = lane%16
template <int NT, bool RELU, bool STATS, int TPW>
__global__ void gemm_wmma(const float* __restrict__ in, const float* __restrict__ W,
                          const float* __restrict__ bias, float* __restrict__ out,
                          int M, int Nout,
                          float* __restrict__ colsum, float* __restrict__ colsumsq) {
  const int K = 64;
  __shared__ float Wf[16 * NT * 32 * 2];   // fragment-ordered weights (<=16KB)

  // ---- stage W into LDS in fragment order ----
  for (int p = threadIdx.x; p < 16 * NT * 32; p += blockDim.x) {
    int lane_p = p & 31;
    int frag   = p >> 5;           // ks*NT + t
    int t_p    = frag % NT;
    int ks_p   = frag / NT;
    int r_p    = lane_p & 15;
    int half_p = lane_p >> 4;
    int krow   = ks_p * 4 + half_p * 2;
    int col    = t_p * 16 + r_p;
    Wf[p * 2 + 0] = W[(long long)krow * Nout + col];
    Wf[p * 2 + 1] = W[(long long)(krow + 1) * Nout + col];
  }
  __syncthreads();
  const v2f* __restrict__ Wfrag = (const v2f*)Wf;  // [ (ks*NT + t)*32 + lane ]

  int wave = threadIdx.x >> 5;
  int wid  = blockIdx.x * (blockDim.x >> 5) + wave;
  int lane = threadIdx.x & 31;
  int r    = lane & 15;   // A: row-in-tile ; B/C: column
  int half = lane >> 4;   // K-pair / M-half selector

  for (int ti = 0; ti < TPW; ++ti) {
    int tile = wid * TPW + ti;
    if (tile * 16 >= M) break;   // wave-uniform

    const float* arow = in + (long long)(tile * 16 + r) * K + half * 2;

    // ---- preload all A fragments for this tile (one load clause) ----
    v2f af[16];
#pragma unroll
    for (int ks = 0; ks < 16; ++ks) {
      af[ks].x = arow[ks * 4];
      af[ks].y = arow[ks * 4 + 1];
    }

    v8f zero = {};
    v8f acc[NT];
#pragma unroll
    for (int t = 0; t < NT; ++t) acc[t] = zero;

#pragma unroll
    for (int ks = 0; ks < 16; ++ks) {
#pragma unroll
      for (int t = 0; t < NT; ++t) {
        v2f b = Wfrag[(ks * NT + t) * 32 + lane];
        acc[t] = __builtin_amdgcn_wmma_f32_16x16x4_f32(
            /*neg_a=*/false, af[ks], /*neg_b=*/false, b,
            /*c_mod=*/(short)0, acc[t], /*reuse_a=*/false, /*reuse_b=*/false);
      }
    }

#pragma unroll
    for (int t = 0; t < NT; ++t) {
      int col  = t * 16 + r;
      float bv = bias[col];
      float s = 0.0f, sq = 0.0f;
#pragma unroll
      for (int v = 0; v < 8; ++v) {
        float val = acc[t][v] + bv;
        if (RELU) val = fmaxf(val, 0.0f);
        int row = tile * 16 + v + half * 8;
        out[(long long)row * Nout + col] = val;
        if (STATS) { s += val; sq += val * val; }
      }
      if (STATS) {
        atomicAdd(&colsum[col], s);
        atomicAdd(&colsumsq[col], sq);
      }
    }
  }
}

// ---------------- BatchNorm ----------------

__global__ void bn_finalize(const float* __restrict__ sum, const float* __restrict__ sumsq,
                            const float* __restrict__ gamma, const float* __restrict__ beta,
                            float* __restrict__ scale, float* __restrict__ shift,
                            int C, float invN) {
  int c = threadIdx.x;
  if (c < C) {
    float mu  = sum[c] * invN;
    float var = sumsq[c] * invN - mu * mu;   // biased variance, matches reference
    float sc  = gamma[c] * rsqrtf(var + BN_EPS);
    scale[c]  = sc;
    shift[c]  = beta[c] - mu * sc;
  }
}

__global__ void bn_apply_relu(const float* __restrict__ h, const float* __restrict__ scale,
                              const float* __restrict__ shift, float* __restrict__ out,
                              long long total, int C) {
  long long i = (long long)blockIdx.x * blockDim.x + threadIdx.x;
  if (i >= total) return;
  int c = (int)(i % C);
  out[i] = fmaxf(h[i] * scale[c] + shift[c], 0.0f);
}

// ---------------- global mean pool ----------------

__global__ void pool_accum(const float* __restrict__ h, const int* __restrict__ batch,
                           float* __restrict__ pool, float* __restrict__ cnt) {
  long long i = (long long)blockIdx.x * blockDim.x + threadIdx.x;
  if (i >= (long long)NN * 32) return;
  int node = (int)(i >> 5);
  int c    = (int)(i & 31);
  int g    = batch[node];
  atomicAdd(&pool[g * 32 + c], h[i]);
  if (c == 0) atomicAdd(&cnt[g], 1.0f);
}

__global__ void pool_final(const float* __restrict__ pool, const float* __restrict__ cnt,
                           float* __restrict__ out) {
  int i = blockIdx.x * blockDim.x + threadIdx.x;
  if (i >= NG * 32) return;
  int g = i >> 5;
  out[i] = pool[i] / fmaxf(cnt[g], 1.0f);
}

// ---------------- host-side layer driver ----------------

static void run_layer(hipStream_t stream,
                      const float* X, const int* srcI, const int* dstI,
                      const float* W1, const float* b1,
                      const float* W2, const float* b2,
                      const float* gamma, const float* beta,
                      float* agg, float* h1, float* h2, float* outb,
                      float* stats, int dout) {
  const int TPB = 256;
  // agg = X  (self term), then scatter-add neighbors
  copy_f4<<<(NN * 64 / 4 + TPB - 1) / TPB, TPB, 0, stream>>>(
      (const float4*)X, (float4*)agg, NN * 64 / 4);
  {
    long long tot = (long long)NE * 16;
    scatter_add<<<(int)((tot + TPB - 1) / TPB), TPB, 0, stream>>>(X, srcI, dstI, agg);
  }
  // 6250 tiles = 125 blocks * 10 waves * 5 tiles/wave (exact)
  // h1 = relu(agg @ W1 + b1)   (K=64, Nout=64)
  gemm_wmma<4, true, false, 5><<<125, 320, 0, stream>>>(agg, W1, b1, h1, NN, 64,
                                                        nullptr, nullptr);
  // zero BN stats (sum[64] | sumsq[64])
  zero_f<<<1, 128, 0, stream>>>(stats, 128);
  float* colsum   = stats;
  float* colsumsq = stats + 64;
  float* scale    = stats + 128;
  float* shift    = stats + 192;
  // h2 = h1 @ W2 + b2, accumulate per-column sum/sumsq
  if (dout == 64) {
    gemm_wmma<4, false, true, 5><<<125, 320, 0, stream>>>(h1, W2, b2, h2, NN, 64,
                                                          colsum, colsumsq);
  } else {
    gemm_wmma<2, false, true, 5><<<125, 320, 0, stream>>>(h1, W2, b2, h2, NN, 32,
                                                          colsum, colsumsq);
  }
  bn_finalize<<<1, 64, 0, stream>>>(colsum, colsumsq, gamma, beta, scale, shift,
                                    dout, 1.0f / (float)NN);
  long long total = (long long)NN * dout;
  bn_apply_relu<<<(int)((total + TPB - 1) / TPB), TPB, 0, stream>>>(
      h2, scale, shift, outb, total, dout);
}

extern "C" void kernel_launch(void* const* d_in, const int* in_sizes, int n_in,
                              void* d_out, int out_size, void* d_ws, size_t ws_size,
                              hipStream_t stream) {
  (void)in_sizes; (void)n_in; (void)out_size; (void)ws_size;

  const float* x     = (const float*)d_in[0];
  const int*   eidx  = (const int*)d_in[1];
  const int*   batch = (const int*)d_in[2];
  const int* srcI = eidx;
  const int* dstI = eidx + NE;

  const float* P[3][6];
  for (int l = 0; l < 3; ++l)
    for (int j = 0; j < 6; ++j)
      P[l][j] = (const float*)d_in[3 + l * 6 + j];   // W1,b1,W2,b2,gamma,beta

  float* bufA  = (float*)d_ws;          // NN*64
  float* bufB  = bufA + (long long)NN * 64;
  float* bufC  = bufB + (long long)NN * 64;
  float* stats = bufC + (long long)NN * 64;  // 256 floats: sum|sumsq|scale|shift
  float* pool  = stats + 256;                // NG*32
  float* cnt   = pool + NG * 32;             // NG

  // Layer 0: X=input; agg=A, h1=B, h2=C, out=A
  run_layer(stream, x, srcI, dstI,
            P[0][0], P[0][1], P[0][2], P[0][3], P[0][4], P[0][5],
            bufA, bufB, bufC, bufA, stats, 64);
  // Layer 1: X=A; agg=B, h1=C, h2=A, out=B
  run_layer(stream, bufA, srcI, dstI,
            P[1][0], P[1][1], P[1][2], P[1][3], P[1][4], P[1][5],
            bufB, bufC, bufA, bufB, stats, 64);
  // Layer 2: X=B; agg=C, h1=A, h2=B, out=C   (dout=32)
  run_layer(stream, bufB, srcI, dstI,
            P[2][0], P[2][1], P[2][2], P[2][3], P[2][4], P[2][5],
            bufC, bufA, bufB, bufC, stats, 32);

  // global mean pool over graphs
  const int TPB = 256;
  zero_f<<<(NG * 32 + NG + TPB - 1) / TPB, TPB, 0, stream>>>(pool, NG * 32 + NG);
  {
    long long tot = (long long)NN * 32;
    pool_accum<<<(int)((tot + TPB - 1) / TPB), TPB, 0, stream>>>(bufC, batch, pool, cnt);
  }
  pool_final<<<(NG * 32 + TPB - 1) / TPB, TPB, 0, stream>>>(pool, cnt, (float*)d_out);
}